// GraphSAGEConv_51135880626287
// MI455X (gfx1250) — compile-verified
//
#include <hip/hip_runtime.h>
#include <hip/hip_bf16.h>

typedef __attribute__((ext_vector_type(2))) float v2f;
typedef __attribute__((ext_vector_type(8))) float v8f;

#define N_NODES 100000
#define N_EDGES 1600000
#define D 64
#define NTILES (N_NODES / 16)   // 6250, exact

// ---------------------------------------------------------------------------
// Phase 0: zero the accumulator workspace (acc[N*64] followed by cnt[N]).
// ---------------------------------------------------------------------------
__global__ void sage_zero(float* __restrict__ p, long long n) {
    long long i = (long long)blockIdx.x * blockDim.x + threadIdx.x;
    const long long stride = (long long)gridDim.x * blockDim.x;
    for (; i < n; i += stride) p[i] = 0.0f;
}

// ---------------------------------------------------------------------------
// Phase 1: edge scatter-add.  16 lanes cooperate on one edge; each lane moves
// a float4 (coalesced 128b gather from x[src], 4 f32 L2 atomics into acc[dst]).
// x (25.6 MB) and acc (25.6 MB) are both resident in the 192 MB L2, so the
// random src/dst indexing is absorbed by L2; HBM sees ~one streaming pass.
// ---------------------------------------------------------------------------
__global__ void sage_scatter(const float* __restrict__ x,
                             const int*  __restrict__ ei,   // [2, N_EDGES]
                             float* __restrict__ acc,
                             float* __restrict__ cnt) {
    const long long tid = (long long)blockIdx.x * blockDim.x + threadIdx.x;
    const long long e   = tid >> 4;
    if (e >= N_EDGES) return;
    const int part = (int)(tid & 15);

    const int src = ei[e];
    const int dst = ei[N_EDGES + e];

    const float4 v = *(const float4*)(x + (size_t)src * D + part * 4);
    float* a = acc + (size_t)dst * D + part * 4;
    atomicAdd(a + 0, v.x);
    atomicAdd(a + 1, v.y);
    atomicAdd(a + 2, v.z);
    atomicAdd(a + 3, v.w);
    if (part == 0) atomicAdd(cnt + dst, 1.0f);
}

// ---------------------------------------------------------------------------
// Phase 2: fused mean + dual GEMM + bias via V_WMMA_F32_16X16X4_F32.
//   out[16x64 tile] = (acc/max(cnt,1)) @ W_l + b_l + x @ W_r
// One wave32 per 16-node tile.  Weights (2 x 16 KB) + bias staged in LDS.
//
// WMMA f32 16x16x4 operand layout (ISA 7.12.2):
//   A (16x4): lane<16 -> M=lane, V0=K0,V1=K1 ; lane>=16 -> M=lane-16, V0=K2,V1=K3
//   B (4x16): V0 -> K=0 (lanes 0-15) / K=2 (lanes 16-31); V1 -> K=1 / K=3
//   C (16x16): VGPR j -> M=j (lanes 0-15) / M=8+j (lanes 16-31), N = lane&15
// ---------------------------------------------------------------------------
__global__ void __launch_bounds__(256)
sage_gemm(const float* __restrict__ x,
          const float* __restrict__ acc,
          const float* __restrict__ cnt,
          const float* __restrict__ Wl,
          const float* __restrict__ bl,
          const float* __restrict__ Wr,
          float* __restrict__ out) {
    __shared__ float sW[2][D * D];
    __shared__ float sB[D];

    for (int i = threadIdx.x; i < D * D; i += blockDim.x) {
        sW[0][i] = Wl[i];
        sW[1][i] = Wr[i];
    }
    if (threadIdx.x < D) sB[threadIdx.x] = bl[threadIdx.x];
    __syncthreads();

    const int wave = threadIdx.x >> 5;
    const int lane = threadIdx.x & 31;
    const int tile = blockIdx.x * (blockDim.x >> 5) + wave;
    if (tile >= NTILES) return;            // uniform per wave -> EXEC stays all-1s

    const int row   = lane & 15;           // M row (both lane halves)
    const int khalf = lane >> 4;           // 0 -> K pair {0,1}; 1 -> K pair {2,3}
    const int node  = tile * 16 + row;

    const float inv = 1.0f / fmaxf(cnt[node], 1.0f);
    const float* aRow = acc + (size_t)node * D;
    const float* xRow = x   + (size_t)node * D;

    // C accumulators for the four N-chunks of 16, pre-loaded with the bias
    // (bias depends only on N = n*16 + row, identical across the 8 M-rows).
    v8f c[4];
#pragma unroll
    for (int n = 0; n < 4; ++n) {
        const float b = sB[n * 16 + row];
#pragma unroll
        for (int j = 0; j < 8; ++j) c[n][j] = b;
    }

#pragma unroll
    for (int kk = 0; kk < 16; ++kk) {      // K = 64, 4 per WMMA step
        const int kbase = kk * 4 + khalf * 2;

        v2f al, ar;
        al.x = aRow[kbase] * inv;  al.y = aRow[kbase + 1] * inv;   // mean path A
        ar.x = xRow[kbase];        ar.y = xRow[kbase + 1];         // root path A

#pragma unroll
        for (int n = 0; n < 4; ++n) {
            const int ncol = n * 16 + row;
            v2f bL, bR;
            bL.x = sW[0][(kbase    ) * D + ncol];
            bL.y = sW[0][(kbase + 1) * D + ncol];
            bR.x = sW[1][(kbase    ) * D + ncol];
            bR.y = sW[1][(kbase + 1) * D + ncol];

            c[n] = __builtin_amdgcn_wmma_f32_16x16x4_f32(
                       false, al, false, bL, (short)0, c[n], false, false);
            c[n] = __builtin_amdgcn_wmma_f32_16x16x4_f32(
                       false, ar, false, bR, (short)0, c[n], false, false);
        }
    }

    // Store: VGPR j of C holds M = 8*khalf + j, column N = n*16 + row.
    const int mbase = tile * 16 + khalf * 8;
#pragma unroll
    for (int n = 0; n < 4; ++n) {
#pragma unroll
        for (int j = 0; j < 8; ++j) {
            out[(size_t)(mbase + j) * D + n * 16 + row] = c[n][j];
        }
    }
}

// ---------------------------------------------------------------------------
extern "C" void kernel_launch(void* const* d_in, const int* in_sizes, int n_in,
                              void* d_out, int out_size, void* d_ws, size_t ws_size,
                              hipStream_t stream) {
    const float* x  = (const float*)d_in[0];   // [N, 64]
    const int*   ei = (const int*)  d_in[1];   // [2, E]
    const float* Wl = (const float*)d_in[2];   // [64, 64]
    const float* bl = (const float*)d_in[3];   // [64]
    const float* Wr = (const float*)d_in[4];   // [64, 64]
    float* out = (float*)d_out;

    float* acc = (float*)d_ws;                       // N*64 f32 sums
    float* cnt = acc + (size_t)N_NODES * D;          // N f32 degree counts

    const long long zn = (long long)N_NODES * D + N_NODES;
    sage_zero<<<2048, 256, 0, stream>>>(acc, zn);

    const long long sthreads = (long long)N_EDGES * 16;
    sage_scatter<<<(unsigned)((sthreads + 255) / 256), 256, 0, stream>>>(x, ei, acc, cnt);

    sage_gemm<<<(NTILES + 7) / 8, 256, 0, stream>>>(x, acc, cnt, Wl, bl, Wr, out);
}